// GNN_69114613727533
// MI455X (gfx1250) — compile-verified
//
#include <hip/hip_runtime.h>
#include <math.h>

#define NN 100000
#define EE 1600000
#define HH 128
#define MM 256
#define LL 6
#define GG 512
#define EPS_V 1e-5f

typedef __attribute__((ext_vector_type(2))) float v2f;
typedef __attribute__((ext_vector_type(8))) float v8f;

__device__ __forceinline__ float swishf(float x) {
    return x / (1.0f + expf(-x));   // x * sigmoid(x)
}

// ---------------- utility kernels ----------------
__global__ void fill_i32(int* p, int v, int n) {
    int i = blockIdx.x * blockDim.x + threadIdx.x;
    if (i < n) p[i] = v;
}
__global__ void copy_i32(const int* a, int* b, int n) {
    int i = blockIdx.x * blockDim.x + threadIdx.x;
    if (i < n) b[i] = a[i];
}
__global__ void count_deg(const int* __restrict__ dst, int* __restrict__ degi, int n) {
    int i = blockIdx.x * blockDim.x + threadIdx.x;
    if (i < n) atomicAdd(&degi[dst[i]], 1);
}
__global__ void dinv_k(const int* __restrict__ degi, float* __restrict__ dinv, int n) {
    int i = blockIdx.x * blockDim.x + threadIdx.x;
    if (i < n) dinv[i] = rsqrtf((float)degi[i]);   // deg >= 1 (self loop)
}

// exclusive scan over degi -> rowst (3-phase block scan)
__global__ void scan1(const int* __restrict__ deg, int* __restrict__ rowst,
                      int* __restrict__ bsum, int n) {
    __shared__ int sm[256];
    int i = blockIdx.x * 256 + threadIdx.x;
    int v = (i < n) ? deg[i] : 0;
    sm[threadIdx.x] = v; __syncthreads();
    for (int off = 1; off < 256; off <<= 1) {
        int u = (threadIdx.x >= (unsigned)off) ? sm[threadIdx.x - off] : 0;
        __syncthreads();
        sm[threadIdx.x] += u;
        __syncthreads();
    }
    if (i < n) rowst[i] = sm[threadIdx.x] - v;          // exclusive
    if (threadIdx.x == 255) bsum[blockIdx.x] = sm[255]; // block total
}
__global__ void scan2(int* bsum, int nb) {
    int acc = 0;
    for (int b = 0; b < nb; ++b) { int t = bsum[b]; bsum[b] = acc; acc += t; }
}
__global__ void scan3(int* __restrict__ rowst, const int* __restrict__ bsum,
                      int n, int total) {
    int i = blockIdx.x * 256 + threadIdx.x;
    if (i < n) rowst[i] += bsum[i >> 8];
    if (i == 0) rowst[n] = total;
}

__global__ void fill_csr_e(const int* __restrict__ src, const int* __restrict__ dst,
                           const float* __restrict__ dinv, int* __restrict__ cursor,
                           int* __restrict__ csrc, float* __restrict__ csw, int n) {
    int e = blockIdx.x * blockDim.x + threadIdx.x;
    if (e < n) {
        int s = src[e], d = dst[e];
        int slot = atomicAdd(&cursor[d], 1);
        csrc[slot] = s;
        csw[slot] = dinv[s] * dinv[d];
    }
}
__global__ void fill_csr_loop(const float* __restrict__ dinv, int* __restrict__ cursor,
                              int* __restrict__ csrc, float* __restrict__ csw, int n) {
    int i = blockIdx.x * blockDim.x + threadIdx.x;
    if (i < n) {
        int slot = atomicAdd(&cursor[i], 1);
        csrc[slot] = i;
        csw[slot] = dinv[i] * dinv[i];
    }
}

__global__ void gcount_k(const int* __restrict__ batch, int* __restrict__ gcnt, int n) {
    int i = blockIdx.x * blockDim.x + threadIdx.x;
    if (i < n) atomicAdd(&gcnt[batch[i]], 1);
}
__global__ void gscan_k(const int* __restrict__ gcnt, int* __restrict__ gstart) {
    __shared__ int sm[GG];
    int t = threadIdx.x;
    int v = gcnt[t];
    sm[t] = v; __syncthreads();
    for (int off = 1; off < GG; off <<= 1) {
        int u = (t >= off) ? sm[t - off] : 0;
        __syncthreads();
        sm[t] += u;
        __syncthreads();
    }
    gstart[t] = sm[t] - v;
    if (t == GG - 1) gstart[GG] = sm[GG - 1];
}

// ---------------- embedding: h = z_embed[z] + x[:,1:]@extra_w.T + extra_b ----------------
__global__ void embed_k(const float* __restrict__ x, const float* __restrict__ zemb,
                        const float* __restrict__ ew, const float* __restrict__ eb,
                        float* __restrict__ h) {
    int idx = blockIdx.x * 256 + threadIdx.x;       // N*H threads
    int n = idx >> 7, c = idx & (HH - 1);
    const float* xr = x + (size_t)n * 4;
    int z = (int)xr[0];
    float v = zemb[(size_t)z * HH + c]
            + xr[1] * ew[c * 3 + 0] + xr[2] * ew[c * 3 + 1] + xr[3] * ew[c * 3 + 2]
            + eb[c];
    h[idx] = v;
}

// ---------------- fp32 WMMA GEMM, 2x2 register-blocked ----------------
// out[n,j] = f(sum_k A[n,k]*W[j,k] + b[j]) (+res).  One wave computes a 32x32
// output block (2x2 WMMA tiles): per k-step 2 A-frags + 2 B-frags feed 4 WMMAs
// -> 8 FLOP/byte from cache (vs 4 for 1x1 tiling).
// N multiple of 32 (100000 = 3125*32); J,K multiples of 32.
template <int ACT, bool RES>
__global__ void gemm_wmma(const float* __restrict__ A, const float* __restrict__ W,
                          const float* __restrict__ bias, const float* __restrict__ res,
                          float* __restrict__ out, int K, int Jt2) {
    int tile = blockIdx.x * blockDim.y + threadIdx.y;
    int tm = tile / Jt2;
    int tj = tile - tm * Jt2;
    int lane = threadIdx.x;
    int m = lane & 15;           // row within A tile / col within B tile
    int half = lane >> 4;        // k sub-slot selector (K={0,1} vs {2,3})
    const float* arow0 = A + (size_t)(tm * 32 + m) * K;
    const float* arow1 = arow0 + (size_t)16 * K;
    const float* wrow0 = W + (size_t)(tj * 32 + m) * K;   // B[k][col] = W[col][k]
    const float* wrow1 = wrow0 + (size_t)16 * K;
    v8f c00 = {}, c01 = {}, c10 = {}, c11 = {};
    for (int k0 = 0; k0 < K; k0 += 4) {
        int kk = k0 + 2 * half;
        v2f a0, a1, b0, b1;
        a0.x = arow0[kk]; a0.y = arow0[kk + 1];
        a1.x = arow1[kk]; a1.y = arow1[kk + 1];
        b0.x = wrow0[kk]; b0.y = wrow0[kk + 1];
        b1.x = wrow1[kk]; b1.y = wrow1[kk + 1];
        c00 = __builtin_amdgcn_wmma_f32_16x16x4_f32(false, a0, false, b0, (short)0, c00, false, false);
        c01 = __builtin_amdgcn_wmma_f32_16x16x4_f32(false, a0, false, b1, (short)0, c01, false, false);
        c10 = __builtin_amdgcn_wmma_f32_16x16x4_f32(false, a1, false, b0, (short)0, c10, false, false);
        c11 = __builtin_amdgcn_wmma_f32_16x16x4_f32(false, a1, false, b1, (short)0, c11, false, false);
    }
    int J = Jt2 * 32;
    int col0 = tj * 32 + m;
    int col1 = col0 + 16;
    float bj0 = bias ? bias[col0] : 0.0f;
    float bj1 = bias ? bias[col1] : 0.0f;
#pragma unroll
    for (int r = 0; r < 8; ++r) {
        int row0 = tm * 32 + r + 8 * half;   // C/D layout: VGPR r -> M=r (+8 for upper lanes)
        int row1 = row0 + 16;
        float v00 = c00[r] + bj0;
        float v01 = c01[r] + bj1;
        float v10 = c10[r] + bj0;
        float v11 = c11[r] + bj1;
        if (ACT) { v00 = swishf(v00); v01 = swishf(v01); v10 = swishf(v10); v11 = swishf(v11); }
        if (RES) {
            v00 += res[(size_t)row0 * J + col0];
            v01 += res[(size_t)row0 * J + col1];
            v10 += res[(size_t)row1 * J + col0];
            v11 += res[(size_t)row1 * J + col1];
        }
        out[(size_t)row0 * J + col0] = v00;
        out[(size_t)row0 * J + col1] = v01;
        out[(size_t)row1 * J + col0] = v10;
        out[(size_t)row1 * J + col1] = v11;
    }
}

// ---------------- CSR gather aggregation: agg[n] = conv_b + sum_{e in CSR(n)} w_e * xw[src_e] ----------------
__global__ void aggregate_k(const float* __restrict__ xw, const int* __restrict__ rowst,
                            const int* __restrict__ csrc, const float* __restrict__ csw,
                            const float* __restrict__ convb, float* __restrict__ agg) {
    int node = blockIdx.x * 8 + threadIdx.y;     // 8 waves per block, 1 node per wave
    int lane = threadIdx.x;                      // 4 channels per lane
    float4 acc = ((const float4*)convb)[lane];
    int s0 = rowst[node], s1 = rowst[node + 1];
    for (int p = s0; p < s1; ++p) {
        int s = csrc[p];
        float w = csw[p];
        float4 v = ((const float4*)(xw + (size_t)s * HH))[lane];
        acc.x += w * v.x; acc.y += w * v.y; acc.z += w * v.z; acc.w += w * v.w;
    }
    ((float4*)(agg + (size_t)node * HH))[lane] = acc;
}

// ---------------- GraphNorm + swish (in-place on agg), one block per graph ----------------
__global__ void graphnorm_k(float* __restrict__ agg, const int* __restrict__ gstart,
                            const float* __restrict__ gamma, const float* __restrict__ beta,
                            const float* __restrict__ msc) {
    __shared__ float red[256];
    __shared__ float meanv[HH];
    __shared__ float rstdv[HH];
    int g = blockIdx.x;
    int t = threadIdx.x;
    int ch = t & (HH - 1);
    int which = t >> 7;                 // 2 threads per channel
    int s = gstart[g], e = gstart[g + 1];
    float cntf = fmaxf((float)(e - s), 1.0f);

    float sum = 0.0f;
    for (int n = s + which; n < e; n += 2) sum += agg[(size_t)n * HH + ch];
    red[t] = sum; __syncthreads();
    if (which == 0) meanv[ch] = (red[ch] + red[ch + 128]) / cntf;
    __syncthreads();

    float mn = msc[ch] * meanv[ch];
    float sq = 0.0f;
    for (int n = s + which; n < e; n += 2) {
        float d = agg[(size_t)n * HH + ch] - mn;
        sq += d * d;
    }
    red[t] = sq; __syncthreads();
    if (which == 0) rstdv[ch] = rsqrtf((red[ch] + red[ch + 128]) / cntf + EPS_V);
    __syncthreads();

    float rs = rstdv[ch], gm = gamma[ch], bt = beta[ch];
    for (int n = s + which; n < e; n += 2) {
        size_t o = (size_t)n * HH + ch;
        float d = agg[o] - mn;
        agg[o] = swishf(gm * d * rs + bt);
    }
}

// ---------------- host orchestration ----------------
extern "C" void kernel_launch(void* const* d_in, const int* in_sizes, int n_in,
                              void* d_out, int out_size, void* d_ws, size_t ws_size,
                              hipStream_t stream) {
    const float* x     = (const float*)d_in[0];
    const int*   ei    = (const int*)d_in[1];
    const int*   batch = (const int*)d_in[2];
    const float* zemb  = (const float*)d_in[3];
    const float* ew    = (const float*)d_in[4];
    const float* eb    = (const float*)d_in[5];
    const float* convw = (const float*)d_in[6];
    const float* convb = (const float*)d_in[7];
    const float* gamma = (const float*)d_in[8];
    const float* beta  = (const float*)d_in[9];
    const float* msc   = (const float*)d_in[10];
    const float* w1    = (const float*)d_in[11];
    const float* b1    = (const float*)d_in[12];
    const float* w2    = (const float*)d_in[13];
    const float* b2    = (const float*)d_in[14];
    float* h = (float*)d_out;                         // h lives in d_out throughout

    // workspace carve (~170 MB)
    float* xwt    = (float*)d_ws;                     // N*M floats (xw uses first N*H, t uses all)
    float* agg    = xwt + (size_t)NN * MM;            // N*H
    float* dinv   = agg + (size_t)NN * HH;            // N
    int*   degi   = (int*)(dinv + NN);                // N
    int*   rowst  = degi + NN;                        // N+1
    int*   cursor = rowst + NN + 1;                   // N
    int*   csrc   = cursor + NN;                      // E+N
    float* csw    = (float*)(csrc + EE + NN);         // E+N
    int*   bsum   = (int*)(csw + EE + NN);            // scan block sums
    int*   gcnt   = bsum + 512;                       // G
    int*   gstart = gcnt + GG;                        // G+1

    const int* src = ei;
    const int* dst = ei + EE;
    int nbN = (NN + 255) / 256;                       // 391
    int nbE = (EE + 255) / 256;

    // graph-static prep (recomputed every call for determinism)
    fill_i32<<<nbN, 256, 0, stream>>>(degi, 1, NN);   // self loop counts as 1
    count_deg<<<nbE, 256, 0, stream>>>(dst, degi, EE);
    scan1<<<nbN, 256, 0, stream>>>(degi, rowst, bsum, NN);
    scan2<<<1, 1, 0, stream>>>(bsum, nbN);
    scan3<<<nbN, 256, 0, stream>>>(rowst, bsum, NN, EE + NN);
    dinv_k<<<nbN, 256, 0, stream>>>(degi, dinv, NN);
    copy_i32<<<nbN, 256, 0, stream>>>(rowst, cursor, NN);
    fill_csr_e<<<nbE, 256, 0, stream>>>(src, dst, dinv, cursor, csrc, csw, EE);
    fill_csr_loop<<<nbN, 256, 0, stream>>>(dinv, cursor, csrc, csw, NN);
    fill_i32<<<2, 256, 0, stream>>>(gcnt, 0, GG);
    gcount_k<<<nbN, 256, 0, stream>>>(batch, gcnt, NN);
    gscan_k<<<1, GG, 0, stream>>>(gcnt, gstart);

    // input embedding
    embed_k<<<(NN * HH) / 256, 256, 0, stream>>>(x, zemb, ew, eb, h);

    dim3 gblk(32, 4);  // 4 waves per block, one 32x32 output block each
    for (int i = 0; i < LL; ++i) {
        // xw = h @ conv_w[i]^T   (bias added in aggregation)
        gemm_wmma<0, false><<<(NN / 32) * (HH / 32) / 4, gblk, 0, stream>>>(
            h, convw + (size_t)i * HH * HH, nullptr, nullptr, xwt, HH, HH / 32);
        // agg = conv_b + CSR-gather(norm * xw[src])
        aggregate_k<<<NN / 8, dim3(32, 8), 0, stream>>>(xwt, rowst, csrc, csw,
                                                        convb + (size_t)i * HH, agg);
        // GraphNorm + swish (in place -> ha)
        graphnorm_k<<<GG, 256, 0, stream>>>(agg, gstart, gamma + (size_t)i * HH,
                                            beta + (size_t)i * HH, msc + (size_t)i * HH);
        // t = swish(ha @ w1^T + b1)
        gemm_wmma<1, false><<<(NN / 32) * (MM / 32) / 4, gblk, 0, stream>>>(
            agg, w1 + (size_t)i * MM * HH, b1 + (size_t)i * MM, nullptr, xwt, HH, MM / 32);
        // h = swish(t @ w2^T + b2) + h   (in-place residual, elementwise-safe)
        gemm_wmma<1, true><<<(NN / 32) * (HH / 32) / 4, gblk, 0, stream>>>(
            xwt, w2 + (size_t)i * HH * MM, b2 + (size_t)i * HH, h, h, MM, HH / 32);
    }
}